// InterpolativeUpsamplerReference_77257871720543
// MI455X (gfx1250) — compile-verified
//
#include <hip/hip_runtime.h>
#include <stdint.h>

// Problem dims (fixed by the reference): x (8,64,256,256) f32 -> out (8,64,511,511) f32
#define H   256
#define W   256
#define OH  511
#define OW  511

// Tiling: 32x32 input tile -> 64x64 output tile per 256-thread block (8 waves)
#define TIN   32
#define HALO  34          // TIN + 2 (reflect halo of 1 each side)
#define XST   40          // LDS row stride for input tile; +3 col shift keeps interior 16B aligned
#define VST   34          // LDS row stride for vertical-pass intermediate

// Types matching the async-to-LDS builtin prototypes (probe-confirmed via clang diagnostics):
//   b128 arg0: vector_size(16) int, AS1, non-const ; b32 arg0: int, AS1, non-const
typedef int v4i __attribute__((vector_size(16)));
typedef __attribute__((address_space(1))) v4i gv4i;
typedef __attribute__((address_space(3))) v4i lv4i;
typedef __attribute__((address_space(1))) int gi32;
typedef __attribute__((address_space(3))) int li32;

// generic->LDS(3): low 32 bits of the flat LDS address are the LDS byte offset
static __device__ __forceinline__ uint32_t lds_off(const void* p) {
    return (uint32_t)(uintptr_t)p;
}

static __device__ __forceinline__ void async_copy_b128(float* dst_lds, const float* src) {
#if __has_builtin(__builtin_amdgcn_global_load_async_to_lds_b128)
    __builtin_amdgcn_global_load_async_to_lds_b128(
        (gv4i*)(uintptr_t)src,
        (lv4i*)(uintptr_t)lds_off(dst_lds),
        0 /*offset*/, 0 /*cpol*/);
#else
    asm volatile("global_load_async_to_lds_b128 %0, %1, off"
                 :: "v"(lds_off(dst_lds)), "v"(src) : "memory");
#endif
}

static __device__ __forceinline__ void async_copy_b32(float* dst_lds, const float* src) {
#if __has_builtin(__builtin_amdgcn_global_load_async_to_lds_b32)
    __builtin_amdgcn_global_load_async_to_lds_b32(
        (gi32*)(uintptr_t)src,
        (li32*)(uintptr_t)lds_off(dst_lds),
        0 /*offset*/, 0 /*cpol*/);
#else
    asm volatile("global_load_async_to_lds_b32 %0, %1, off"
                 :: "v"(lds_off(dst_lds)), "v"(src) : "memory");
#endif
}

static __device__ __forceinline__ void wait_async0() {
#if __has_builtin(__builtin_amdgcn_s_wait_asynccnt)
    __builtin_amdgcn_s_wait_asynccnt(0);
#else
    asm volatile("s_wait_asynccnt 0x0" ::: "memory");
#endif
}

static __device__ __forceinline__ int refl(int i) {
    // jnp reflect mode: mirror without repeating the edge sample
    i = (i < 0) ? -i : i;
    return (i >= H) ? (2 * H - 2 - i) : i;
}

// Per axis (reflect at edges):
//   out[2m]   = (x[m-1] + 6 x[m] + x[m+1]) / 8
//   out[2m+1] = (x[m] + x[m+1]) / 2
__global__ __launch_bounds__(256)
void upsample2x_1331_kernel(const float* __restrict__ x, float* __restrict__ out) {
    __shared__ float Xs[HALO * XST];   // 34 x (40) reflect-padded input tile, cols at +3
    __shared__ float Vs[64 * VST];     // 64 output rows x 34 halo cols (vertical pass)

    const int tid = threadIdx.x;
    const int tx = blockIdx.x, ty = blockIdx.y, bc = blockIdx.z;
    const float* __restrict__ xin = x + (size_t)bc * (H * W);

    const int gc0 = tx * TIN;          // first interior global column of this tile

    // ---- stage interior via async DMA: 34 rows x 8 float4 (cols gc0..gc0+31, never col-reflected)
#pragma unroll
    for (int k = 0; k < 2; ++k) {
        const int i = tid + (k << 8);
        if (i < HALO * 8) {
            const int r = i >> 3, q = i & 7;
            const int gr = refl(ty * TIN + r - 1);
            async_copy_b128(&Xs[r * XST + 4 + q * 4], xin + gr * W + gc0 + q * 4);
        }
    }
    // ---- stage the two reflected halo columns: 34 rows x 2 scalars
    if (tid < HALO * 2) {
        const int r = tid >> 1, s = tid & 1;
        const int gr = refl(ty * TIN + r - 1);
        const int gc = refl(s ? (gc0 + TIN) : (gc0 - 1));
        async_copy_b32(&Xs[r * XST + 3 + (s ? 33 : 0)], xin + gr * W + gc);
    }
    wait_async0();
    __syncthreads();

    // ---- vertical pass: 32 m-rows x 34 halo cols -> Vs[64][34], row scales folded in
#pragma unroll
    for (int k = 0; k < 5; ++k) {
        const int i = tid + (k << 8);           // k<4 always in range (tid+768 < 1088)
        if (i < TIN * HALO) {
            const int m = i / HALO;             // 0..31
            const int j = i - m * HALO;         // 0..33
            const float a = Xs[(m + 0) * XST + 3 + j];
            const float b = Xs[(m + 1) * XST + 3 + j];
            const float c = Xs[(m + 2) * XST + 3 + j];
            Vs[(2 * m + 0) * VST + j] = 0.125f * (a + 6.0f * b + c);
            Vs[(2 * m + 1) * VST + j] = 0.5f   * (b + c);
        }
    }
    __syncthreads();

    // ---- horizontal pass + streaming stores ----
    // One wave covers one full 64-col tile row; lane l owns output pair (cols 2l, 2l+1),
    // reading a,b,c from Vs once and producing both outputs (halves LDS traffic).
    const int lane = tid & 31;
    const int wv   = tid >> 5;                 // 0..7
    const int oxe  = tx * 64 + 2 * lane;       // even col: 2l <= 510, always valid
    const bool oddok = (oxe + 1) < OW;         // only fails for tx==7, lane==31
    float* __restrict__ op = out + (size_t)bc * ((size_t)OH * OW);

#pragma unroll
    for (int rr = 0; rr < 8; ++rr) {
        const int oy = wv + 8 * rr;            // 0..63
        const float a = Vs[oy * VST + lane + 0];
        const float b = Vs[oy * VST + lane + 1];
        const float c = Vs[oy * VST + lane + 2];
        const float ve = 0.125f * (a + 6.0f * b + c);
        const float vo = 0.5f   * (b + c);
        const int oyg = ty * 64 + oy;
        if (oyg < OH) {                        // wave-uniform guard
            float* rowp = op + (size_t)oyg * OW + oxe;
            // 535 MB streaming output > 192 MB L2: non-temporal
            __builtin_nontemporal_store(ve, rowp);
            if (oddok) __builtin_nontemporal_store(vo, rowp + 1);
        }
    }
}

extern "C" void kernel_launch(void* const* d_in, const int* in_sizes, int n_in,
                              void* d_out, int out_size, void* d_ws, size_t ws_size,
                              hipStream_t stream) {
    (void)n_in; (void)out_size; (void)d_ws; (void)ws_size;
    const float* x = (const float*)d_in[0];
    float* out = (float*)d_out;
    const int BC = in_sizes[0] / (H * W);          // 8*64 = 512 images
    dim3 grid(W / TIN, H / TIN, BC);               // (8, 8, 512)
    upsample2x_1331_kernel<<<grid, dim3(256), 0, stream>>>(x, out);
}